// spiral_conv_19945828123195
// MI455X (gfx1250) — compile-verified
//
#include <hip/hip_runtime.h>

typedef __attribute__((ext_vector_type(2))) float v2f;
typedef __attribute__((ext_vector_type(8))) float v8f;

#define B_    8
#define N_    50000
#define FIN   32
#define L_    16
#define FOUT  32
#define K_    (L_ * FIN)   // 512
#define PAD   516          // LDS row stride in floats; 516 % 64 == 4 -> conflict-free B reads

// Workgroup: 256 threads = 8 wave32s. Wave w handles batch w; block handles a
// 16-node tile (grid = N/16 = 3125). Each wave does a [16 x 512] x [512 x 32]
// GEMM tile via V_WMMA_F32_16X16X4_F32 (exact f32, matches reference).
__global__ __launch_bounds__(256) void spiral_conv_wmma(
    const float* __restrict__ x,      // [B, N, FIN]
    const float* __restrict__ w,      // [K_, FOUT]
    const int*   __restrict__ idx,    // [N * L_]
    float*       __restrict__ out)    // [B, N, FOUT]
{
    extern __shared__ float wt[];     // transposed filter: wt[n * PAD + k] = w[k][n]

    const int tid  = threadIdx.x;
    const int lane = tid & 31;
    const int wave = tid >> 5;                 // == batch index (8 waves = 8 batches)
    const int m0   = blockIdx.x * 16;          // node tile base

    // ---- Stage W transposed into LDS (coalesced global reads) ----
    {
        const int n = tid & 31;                // lane n reads column n
        for (int k = (tid >> 5); k < K_; k += 8) {
            wt[n * PAD + k] = w[k * FOUT + n];
        }
    }
    __syncthreads();

    const int row   = lane & 15;               // node row handled by this lane
    const int khalf = (lane >> 4) * 2;         // A/B fragment K offset: 0 or 2

    const float* xb = x + (size_t)wave * N_ * FIN;

    // Per-lane LDS base pointers for the two B column tiles (cols 0-15 / 16-31)
    const float* wt0 = &wt[(row)      * PAD + khalf];
    const float* wt1 = &wt[(row + 16) * PAD + khalf];

    v8f acc0 = {};   // output cols  0..15
    v8f acc1 = {};   // output cols 16..31

    const int* idxrow = idx + (size_t)(m0 + row) * L_;
    int nb_next = idxrow[0];

    for (int l = 0; l < L_; ++l) {
        const int nb = nb_next;
        if (l + 1 < L_) {
            nb_next = idxrow[l + 1];
            // gfx1250 global_prefetch_b8: warm the next gathered row
            __builtin_prefetch(xb + (size_t)nb_next * FIN, 0, 1);
        }
        const float* arow = xb + (size_t)nb * FIN + khalf;
        const int    kb   = l * FIN;
#pragma unroll
        for (int kk = 0; kk < FIN; kk += 4) {
            // A frag: lanes 0-15 -> A[row][k0..k0+1], lanes 16-31 -> A[row][k0+2..k0+3]
            v2f a  = *(const v2f*)(arow + kk);
            // B frags from LDS: VGPR0 <- W[k0+khalf][col], VGPR1 <- W[k0+khalf+1][col]
            v2f b0 = *(const v2f*)(wt0 + kb + kk);
            v2f b1 = *(const v2f*)(wt1 + kb + kk);
            acc0 = __builtin_amdgcn_wmma_f32_16x16x4_f32(
                false, a, false, b0, (short)0, acc0, false, false);
            acc1 = __builtin_amdgcn_wmma_f32_16x16x4_f32(
                false, a, false, b1, (short)0, acc1, false, false);
        }
    }

    // ---- Store D: VGPR r -> (row = r + 8*(lane>=16), col = lane&15) ----
    const int crow = (lane >> 4) * 8;
    const int ccol = lane & 15;
    float* ob = out + ((size_t)wave * N_ + (m0 + crow)) * FOUT + ccol;
#pragma unroll
    for (int r = 0; r < 8; ++r) {
        ob[(size_t)r * FOUT]      = acc0[r];
        ob[(size_t)r * FOUT + 16] = acc1[r];
    }
}

extern "C" void kernel_launch(void* const* d_in, const int* in_sizes, int n_in,
                              void* d_out, int out_size, void* d_ws, size_t ws_size,
                              hipStream_t stream) {
    const float* x   = (const float*)d_in[0];
    const float* w   = (const float*)d_in[1];
    const int*   idx = (const int*)d_in[2];
    float*       out = (float*)d_out;

    const size_t shmem = (size_t)FOUT * PAD * sizeof(float);  // ~66 KB (<< 320 KB/WGP)
    spiral_conv_wmma<<<dim3(N_ / 16), dim3(256), shmem, stream>>>(x, w, idx, out);
}